// BiTemperedLogisticLoss_20787641713071
// MI455X (gfx1250) — compile-verified
//
#include <hip/hip_runtime.h>
#include <cstdint>
#include <cmath>

#define C_DIM 1000
#define ROWS_PER_BLOCK 8
#define KPL 32  // elements per lane: 31 full strided chunks + 8-lane tail

__device__ __forceinline__ float wave_sum(float v) {
#pragma unroll
    for (int off = 16; off > 0; off >>= 1) v += __shfl_xor(v, off, 32);
    return v;
}
__device__ __forceinline__ float wave_max(float v) {
#pragma unroll
    for (int off = 16; off > 0; off >>= 1) v = fmaxf(v, __shfl_xor(v, off, 32));
    return v;
}

// One wave (wave32) per row. Row logits live in 32 VGPRs/lane for all 21
// partition-function passes. Targets row is staged LDS via CDNA5 async copy,
// overlapped with the fixed-point iterations.
__global__ __launch_bounds__(256)
void bitempered_rows(const float* __restrict__ inp, const float* __restrict__ tgt,
                     float* __restrict__ partial, int N,
                     float t_on, float t_off, float K_on, float K_off) {
    __shared__ float s_tgt[ROWS_PER_BLOCK * 1024];  // 1024-float stride per wave
    __shared__ float s_part[ROWS_PER_BLOCK];

    const int lane = threadIdx.x & 31;
    const int wave = threadIdx.x >> 5;
    const int row  = blockIdx.x * ROWS_PER_BLOCK + wave;
    const bool active = row < N;

    float wsum = 0.0f;
    if (active) {
        const float* rowp = inp + (size_t)row * C_DIM;
        const float* trow = tgt + (size_t)row * C_DIM;

        // ---- CDNA5 async copy: targets row -> LDS (overlaps with compute) ----
        unsigned voff  = (unsigned)lane * 4u;
        unsigned laddr = (unsigned)(uintptr_t)(&s_tgt[wave * 1024]) + voff;
#pragma unroll
        for (int k = 0; k < 31; ++k) {
            asm volatile("global_load_async_to_lds_b32 %0, %1, %2 offset:%3"
                         :: "v"(laddr), "v"(voff), "s"(trow), "i"(k * 128)
                         : "memory");
        }
        if (lane < 8) {
            asm volatile("global_load_async_to_lds_b32 %0, %1, %2 offset:3968"
                         :: "v"(laddr), "v"(voff), "s"(trow)
                         : "memory");
        }

        // ---- load logits into registers (read exactly once) ----
        float x[KPL];
#pragma unroll
        for (int k = 0; k < 31; ++k) x[k] = rowp[lane + 32 * k];
        x[31] = (lane < 8) ? rowp[lane + 992] : -INFINITY;  // pad: contributes 0

        // row max, then shift
        float mx = x[0];
#pragma unroll
        for (int k = 1; k < KPL; ++k) mx = fmaxf(mx, x[k]);
        const float mu = wave_max(mx);
#pragma unroll
        for (int k = 0; k < KPL; ++k) x[k] -= mu;

        // 20 fixed-point iterations + final partition sum (21 passes).
        // exp_t(x, 1.2) = (1 - 0.2 x)^(-5) = r^5 with r = rcp(fma(x,-0.2s,1)).
        float s = 1.0f, z = 1.0f;
#pragma unroll 1
        for (int it = 0; it < 21; ++it) {
            const float m = -0.2f * s;
            float acc = 0.0f;
#pragma unroll
            for (int k = 0; k < KPL; ++k) {
                float u  = __builtin_fmaf(x[k], m, 1.0f);   // >= 1, relu never active
                float r  = __builtin_amdgcn_rcpf(u);
                float r2 = r * r;
                float r4 = r2 * r2;
                acc = __builtin_fmaf(r4, r, acc);           // += r^5
            }
            z = wave_sum(acc);
            s = __builtin_amdgcn_exp2f(-0.2f * __builtin_amdgcn_logf(z)); // z^(1-t2)
        }
        // lambda = 5(z^0.2 - 1) + mu  =>  u_c = z^0.2 - 0.2*(a_c - mu)
        const float d = __builtin_amdgcn_exp2f(0.2f * __builtin_amdgcn_logf(z));

        asm volatile("s_wait_asynccnt 0" ::: "memory");      // targets now in LDS

        // loss_c = -t_c*(r4-1)/0.8 + r9/1.8 + K(t_c), with r4=p^0.8, r9=p^1.8
        float lacc = 0.0f;
#pragma unroll
        for (int k = 0; k < KPL; ++k) {
            int   c  = lane + 32 * k;
            float u  = __builtin_fmaf(x[k], -0.2f, d);
            float r  = __builtin_amdgcn_rcpf(u);
            float r2 = r * r;
            float r4 = r2 * r2;
            float r8 = r4 * r4;
            float r9 = r8 * r;
            float tv = s_tgt[wave * 1024 + c];               // c <= 1023: safe
            bool  on = (tv != 0.0f);
            float tc = on ? t_on : t_off;
            float Kc = on ? K_on : K_off;
            float lc = __builtin_fmaf(r9, (1.0f / 1.8f), Kc)
                     - tc * (r4 - 1.0f) * 1.25f;
            if (c < C_DIM) lacc += lc;
        }
        wsum = wave_sum(lacc);
    }

    if (lane == 0) s_part[wave] = active ? wsum : 0.0f;
    __syncthreads();
    if (threadIdx.x == 0) {
        float b = 0.0f;
#pragma unroll
        for (int w = 0; w < ROWS_PER_BLOCK; ++w) b += s_part[w];
        partial[blockIdx.x] = b;
    }
}

__global__ __launch_bounds__(256)
void bitempered_finalize(const float* __restrict__ partial, float* __restrict__ out,
                         int nblocks, float invN) {
    __shared__ float sm[256];
    float a = 0.0f;
    for (int i = threadIdx.x; i < nblocks; i += 256) a += partial[i];
    sm[threadIdx.x] = a;
    __syncthreads();
    for (int off = 128; off > 0; off >>= 1) {
        if ((int)threadIdx.x < off) sm[threadIdx.x] += sm[threadIdx.x + off];
        __syncthreads();
    }
    if (threadIdx.x == 0) out[0] = sm[0] * invN;
}

extern "C" void kernel_launch(void* const* d_in, const int* in_sizes, int n_in,
                              void* d_out, int out_size, void* d_ws, size_t ws_size,
                              hipStream_t stream) {
    const float* inp = (const float*)d_in[0];
    const float* tgt = (const float*)d_in[1];
    float* out = (float*)d_out;
    float* ws  = (float*)d_ws;

    const int C = C_DIM;
    const int N = in_sizes[0] / C;
    const int blocks = (N + ROWS_PER_BLOCK - 1) / ROWS_PER_BLOCK;

    // Smoothed-target constants (targets are exact one-hot, two values only).
    const double LS  = 0.05;
    const double on  = 1.0 - ((double)C / (double)(C - 1)) * LS;
    const double off = LS / (double)(C - 1);
    auto logt = [](double y) { return (pow(y, 0.8) - 1.0) / 0.8; };
    const double Kon  = on  * logt(on  + 1e-8) - pow(on,  1.8) / 1.8;
    const double Koff = off * logt(off + 1e-8) - pow(off, 1.8) / 1.8;

    bitempered_rows<<<blocks, 256, 0, stream>>>(
        inp, tgt, ws, N, (float)on, (float)off, (float)Kon, (float)Koff);
    bitempered_finalize<<<1, 256, 0, stream>>>(ws, out, blocks, 1.0f / (float)N);
}